// EvolutionBank_76836964926215
// MI455X (gfx1250) — compile-verified
//
#include <hip/hip_runtime.h>
#include <stddef.h>

// EvolutionBank: scatter emb rows into a rolling bank, bump pointers, gather windows.
// Pure memory-bound op: ~3.5 GB of traffic, 0 FLOPs -> ~150us floor at 23.3 TB/s.
// Bulk bank copy uses gfx1250 async global<->LDS B128 data movers (ASYNCcnt),
// with NT temporal hints (cpol TH=1) since the 1.5 GB stream is 8x the 192 MB L2.

typedef float f4 __attribute__((ext_vector_type(4)));
typedef int   v4i __attribute__((ext_vector_type(4)));

#if defined(__gfx1250__) &&                                                     \
    __has_builtin(__builtin_amdgcn_global_load_async_to_lds_b128) &&            \
    __has_builtin(__builtin_amdgcn_global_store_async_from_lds_b128) &&         \
    __has_builtin(__builtin_amdgcn_s_wait_asynccnt)
#define EB_ASYNC_LDS 1
// Builtin pointer parameter types (from hipcc diagnostic):
//   arg0: v4i in address space 1 (global), arg1: v4i in address space 3 (LDS)
typedef __attribute__((address_space(1))) v4i gbl_v4i;
typedef __attribute__((address_space(3))) v4i lds_v4i;
// CPol: TH bits [2:0]; TH=1 -> non-temporal for both loads and stores.
#define EB_CPOL_NT 1
#else
#define EB_ASYNC_LDS 0
#endif

#define EB_THREADS 256
#define EB_COPY_ITERS 4

// ---------------------------------------------------------------------------
// Kernel A (main): guard-free bulk copy. Grid exactly covers
// gridDim * 256 * EB_COPY_ITERS float4 elements, so no bounds checks and no
// saveexec branches around the async transfers: 4 async B128 global->LDS
// loads, s_wait_asynccnt 0, 4 async LDS->global stores. Data never touches
// VGPRs. Fallback: plain non-temporal float4 copy.
// ---------------------------------------------------------------------------
__global__ __launch_bounds__(EB_THREADS) void eb_copy_bank_main(
    const f4* __restrict__ src, f4* __restrict__ dst) {
  const long long stride = (long long)gridDim.x * EB_THREADS;
  const long long base = (long long)blockIdx.x * EB_THREADS + threadIdx.x;
#if EB_ASYNC_LDS
  __shared__ f4 stage[EB_COPY_ITERS][EB_THREADS];
#pragma unroll
  for (int k = 0; k < EB_COPY_ITERS; ++k) {
    __builtin_amdgcn_global_load_async_to_lds_b128(
        (gbl_v4i*)(src + (base + (long long)k * stride)),
        (lds_v4i*)&stage[k][threadIdx.x],
        /*offset=*/0, /*cpol=*/EB_CPOL_NT);
  }
  __builtin_amdgcn_s_wait_asynccnt(0);  // loads landed in LDS
#pragma unroll
  for (int k = 0; k < EB_COPY_ITERS; ++k) {
    __builtin_amdgcn_global_store_async_from_lds_b128(
        (gbl_v4i*)(dst + (base + (long long)k * stride)),
        (lds_v4i*)&stage[k][threadIdx.x],
        /*offset=*/0, /*cpol=*/EB_CPOL_NT);
  }
  __builtin_amdgcn_s_wait_asynccnt(0);  // drain stores before wave exit
#else
#pragma unroll
  for (int k = 0; k < EB_COPY_ITERS; ++k) {
    const long long ei = base + (long long)k * stride;
    f4 v = __builtin_nontemporal_load(src + ei);
    __builtin_nontemporal_store(v, dst + ei);
  }
#endif
}

// Kernel A (tail): guarded scalar-f4 copy for the remainder (zero for the
// reference shapes: 96M float4s divide 1024 exactly; kept for generality).
__global__ __launch_bounds__(EB_THREADS) void eb_copy_bank_tail(
    const f4* __restrict__ src, f4* __restrict__ dst, long long lo, long long n16) {
  const long long ei = lo + (long long)blockIdx.x * EB_THREADS + threadIdx.x;
  if (ei < n16) {
    f4 v = __builtin_nontemporal_load(src + ei);
    __builtin_nontemporal_store(v, dst + ei);
  }
}

// ---------------------------------------------------------------------------
// Kernel B: one wave32 per (i, w) output row.
//   p = ptr[idx[i]] % W
//   out[i,w,:]      = (w==p) ? emb[i,:] : bank[idx[i],w,:]
//   new_bank[idx[i],p,:] = emb[i,:]   (overwrites kernel A's copy; stream order
//                                      guarantees A completed first)
// ---------------------------------------------------------------------------
__global__ __launch_bounds__(EB_THREADS) void eb_scatter_gather(
    const float* __restrict__ bank, const float* __restrict__ emb,
    const int* __restrict__ ptr, const int* __restrict__ idx,
    float* __restrict__ out, float* __restrict__ nbank,
    int Bn, int W, int D) {
  const int gid = blockIdx.x * EB_THREADS + threadIdx.x;
  const int lane = gid & 31;
  const int row = gid >> 5;
  if (row >= Bn * W) return;
  const int i = row / W;
  const int w = row - i * W;
  const int node = idx[i];
  const int pv = ptr[node];
  const int p = ((pv % W) + W) % W;  // jnp.mod semantics (ptr >= 0 anyway)

  const int nf4 = D >> 2;  // 128-bit chunks per row (D is a multiple of 4)
  const bool is_scatter = (w == p);
  const f4* src = is_scatter
                      ? (const f4*)(emb + (size_t)i * D)
                      : (const f4*)(bank + ((size_t)node * W + w) * D);
  f4* dst_out = (f4*)(out + (size_t)row * D);
  f4* dst_bank = (f4*)(nbank + ((size_t)node * W + w) * D);

  for (int c = lane; c < nf4; c += 32) {
    f4 v = __builtin_nontemporal_load(src + c);
    __builtin_nontemporal_store(v, dst_out + c);
    if (is_scatter) __builtin_nontemporal_store(v, dst_bank + c);
  }
}

// ---------------------------------------------------------------------------
// Kernels C/D: new_ptr = ptr; new_ptr[idx] = ptr[idx] + 1 (idx unique).
// new_ptr lives in the float-typed d_out buffer; int bits are written raw.
// ---------------------------------------------------------------------------
__global__ __launch_bounds__(EB_THREADS) void eb_ptr_copy(
    const int* __restrict__ ptr, int* __restrict__ nptr, int n) {
  const int g = blockIdx.x * EB_THREADS + threadIdx.x;
  if (g < n) nptr[g] = ptr[g];
}

__global__ __launch_bounds__(EB_THREADS) void eb_ptr_inc(
    const int* __restrict__ ptr, const int* __restrict__ idx,
    int* __restrict__ nptr, int Bn) {
  const int g = blockIdx.x * EB_THREADS + threadIdx.x;
  if (g < Bn) {
    const int node = idx[g];
    nptr[node] = ptr[node] + 1;
  }
}

extern "C" void kernel_launch(void* const* d_in, const int* in_sizes, int n_in,
                              void* d_out, int out_size, void* d_ws, size_t ws_size,
                              hipStream_t stream) {
  const float* bank = (const float*)d_in[0];
  const float* emb  = (const float*)d_in[1];
  const int*   ptr  = (const int*)d_in[2];
  const int*   idx  = (const int*)d_in[3];

  const long long bank_elems = (long long)in_sizes[0];  // N*W*D
  const int N  = in_sizes[2];                           // num nodes
  const int Bn = in_sizes[3];                           // batch
  const int D  = in_sizes[1] / Bn;                      // dim
  const int W  = (int)(bank_elems / ((long long)N * D));

  float* out   = (float*)d_out;                         // (Bn, W, D)
  float* nbank = out + (size_t)Bn * W * D;              // (N, W, D)
  int*   nptr  = (int*)(nbank + (size_t)N * W * D);     // (N,) int bits

  // A: bulk copy bank -> new_bank (guard-free main grid + optional tail)
  const long long n16 = bank_elems / 4;                 // float4 count
  const long long per_blk = (long long)EB_THREADS * EB_COPY_ITERS;  // 1024
  const long long blocksA = n16 / per_blk;              // full blocks
  const long long mainA = blocksA * per_blk;            // elements covered
  if (blocksA > 0)
    eb_copy_bank_main<<<(int)blocksA, EB_THREADS, 0, stream>>>((const f4*)bank,
                                                               (f4*)nbank);
  const long long tailA = n16 - mainA;
  if (tailA > 0)
    eb_copy_bank_tail<<<(int)((tailA + EB_THREADS - 1) / EB_THREADS), EB_THREADS,
                        0, stream>>>((const f4*)bank, (f4*)nbank, mainA, n16);

  // B: scatter emb rows + gather out windows (after A on the same stream)
  const long long rows = (long long)Bn * W;
  const int blocksB = (int)((rows * 32 + EB_THREADS - 1) / EB_THREADS);
  eb_scatter_gather<<<blocksB, EB_THREADS, 0, stream>>>(bank, emb, ptr, idx,
                                                        out, nbank, Bn, W, D);

  // C/D: pointer copy then increment (D after C on the same stream)
  eb_ptr_copy<<<(N + EB_THREADS - 1) / EB_THREADS, EB_THREADS, 0, stream>>>(ptr, nptr, N);
  eb_ptr_inc<<<(Bn + EB_THREADS - 1) / EB_THREADS, EB_THREADS, 0, stream>>>(ptr, idx, nptr, Bn);
}